// BaseType_69535520522293
// MI455X (gfx1250) — compile-verified
//
#include <hip/hip_runtime.h>
#include <hip/hip_bf16.h>
#include <math.h>

// ---------------- Sizes (from reference) ----------------
#define Vv   50257
#define Ee   512
#define Hh   1024
#define Tt   512
#define Ss   8
#define TSs  128
#define OOV  20
#define VPO  (Vv + OOV)        // 50277
#define COMB (3*Hh + Ee)       // 3584

// ---------------- Workspace layout (float offsets) ----------------
#define WS_Q       0            // 1024
#define WS_SC_ENC  1024         // 512
#define WS_SC_NAME 1536         // 1024
#define WS_SC_TYPE 2560         // 1024
#define WS_C_ENC   3584         // 1024
#define WS_C_NAME  4608         // 1024
#define WS_C_TYPE  5632         // 1024
#define WS_X       6656         // 512
#define WS_GATES   7168         // 4096
#define WS_H1      11264        // 1024
#define WS_C1      12288        // 1024
#define WS_QC      13312        // 1024
#define WS_ATTN    14336        // 1024 (S*TS)
#define WS_SIM     15360        // 8
#define WS_GENP    15368        // 1
#define WS_MS      15370        // 2 (max, inv_sum)
#define WS_PCOPY   15376        // 50277
#define WS_LOGITS  65664        // 50257

typedef __attribute__((ext_vector_type(2))) float v2f;
typedef __attribute__((ext_vector_type(8))) float v8f;

// ---------------- Reductions (wave32) ----------------
__device__ __forceinline__ float wredSum(float v) {
#pragma unroll
  for (int o = 16; o > 0; o >>= 1) v += __shfl_down(v, o, 32);
  return v;
}
__device__ __forceinline__ float wredMax(float v) {
#pragma unroll
  for (int o = 16; o > 0; o >>= 1) v = fmaxf(v, __shfl_down(v, o, 32));
  return v;
}
__device__ float blockSum(float v) {
  __shared__ float sh[32];
  __syncthreads();
  int lane = threadIdx.x & 31, wid = threadIdx.x >> 5;
  v = wredSum(v);
  if (lane == 0) sh[wid] = v;
  __syncthreads();
  int nw = (blockDim.x + 31) >> 5;
  float r = (threadIdx.x < nw) ? sh[threadIdx.x] : 0.f;
  if (wid == 0) r = wredSum(r);
  if (threadIdx.x == 0) sh[0] = r;
  __syncthreads();
  return sh[0];
}
__device__ float blockMax(float v) {
  __shared__ float sh[32];
  __syncthreads();
  int lane = threadIdx.x & 31, wid = threadIdx.x >> 5;
  v = wredMax(v);
  if (lane == 0) sh[wid] = v;
  __syncthreads();
  int nw = (blockDim.x + 31) >> 5;
  float r = (threadIdx.x < nw) ? sh[threadIdx.x] : -INFINITY;
  if (wid == 0) r = wredMax(r);
  if (threadIdx.x == 0) sh[0] = r;
  __syncthreads();
  return sh[0];
}

// ---------------- Small kernels ----------------
__global__ void k_zero(float* p, int n) {
  int i = blockIdx.x * blockDim.x + threadIdx.x;
  if (i < n) p[i] = 0.f;
}

// y[row] = dot(A[baseOff + row*rowPitch, :cols], x) (+ b[row]) ; one wave per row.
// cols must be a multiple of 128 (all uses: 512 or 1024).
__global__ void k_gemv(const float* __restrict__ A, const float* __restrict__ x,
                       const float* __restrict__ b, float* __restrict__ y,
                       int rows, int cols, long long rowPitch, long long baseOff) {
  int wpb = blockDim.x >> 5;
  int row = blockIdx.x * wpb + (threadIdx.x >> 5);
  int lane = threadIdx.x & 31;
  if (row >= rows) return;
  const float* a = A + baseOff + (long long)row * rowPitch;
  float s = 0.f;
  for (int k = lane * 4; k < cols; k += 128) {
    float4 va = *(const float4*)(a + k);
    float4 vx = *(const float4*)(x + k);
    s = fmaf(va.x, vx.x, fmaf(va.y, vx.y, fmaf(va.z, vx.z, fmaf(va.w, vx.w, s))));
  }
  s = wredSum(s);
  if (lane == 0) y[row] = b ? (s + b[row]) : s;
}

// in-place softmax over v[0..n), n <= blockDim.x
__global__ void k_softmax(float* v, int n) {
  int tid = threadIdx.x;
  float val = (tid < n) ? v[tid] : -INFINITY;
  float m = blockMax(val);
  float e = (tid < n) ? __expf(val - m) : 0.f;
  float s = blockSum(e);
  if (tid < n) v[tid] = e / s;
}

// c[j] = sum_t s[t] * M[t*cols + j]
__global__ void k_colsum(const float* __restrict__ M, const float* __restrict__ s,
                         float* __restrict__ c, int T, int cols) {
  int j = blockIdx.x * blockDim.x + threadIdx.x;
  if (j >= cols) return;
  float acc = 0.f;
  for (int t = 0; t < T; ++t) acc = fmaf(s[t], M[(long long)t * cols + j], acc);
  c[j] = acc;
}

// x[row] = tanh( W_ma_comb[row,:] . concat(e, c_enc, c_name, c_type) + bias[row] )
__global__ void k_x(const float* __restrict__ W, const float* __restrict__ bias,
                    const float* __restrict__ e, const float* __restrict__ ce,
                    const float* __restrict__ cn, const float* __restrict__ ct,
                    float* __restrict__ x) {
  int row = blockIdx.x * (blockDim.x >> 5) + (threadIdx.x >> 5);
  int lane = threadIdx.x & 31;
  if (row >= Ee) return;
  const float* w = W + (long long)row * COMB;
  float s = 0.f;
  for (int k = lane; k < COMB; k += 32) {
    float cv = (k < Ee) ? e[k]
             : (k < Ee + Hh) ? ce[k - Ee]
             : (k < Ee + 2 * Hh) ? cn[k - Ee - Hh]
             : ct[k - Ee - 2 * Hh];
    s = fmaf(w[k], cv, s);
  }
  s = wredSum(s);
  if (lane == 0) x[row] = tanhf(s + bias[row]);
}

// gates[row] = W_ih[row,:].x + W_hh[row,:].h0 + b_ih[row] + b_hh[row]
__global__ void k_gates(const float* __restrict__ Wih, const float* __restrict__ Whh,
                        const float* __restrict__ bih, const float* __restrict__ bhh,
                        const float* __restrict__ x, const float* __restrict__ h0,
                        float* __restrict__ gates) {
  int row = blockIdx.x * (blockDim.x >> 5) + (threadIdx.x >> 5);
  int lane = threadIdx.x & 31;
  if (row >= 4 * Hh) return;
  const float* wi = Wih + (long long)row * Ee;
  const float* wh = Whh + (long long)row * Hh;
  float s = 0.f;
  for (int k = lane * 4; k < Ee; k += 128) {
    float4 a = *(const float4*)(wi + k); float4 v = *(const float4*)(x + k);
    s = fmaf(a.x, v.x, fmaf(a.y, v.y, fmaf(a.z, v.z, fmaf(a.w, v.w, s))));
  }
  for (int k = lane * 4; k < Hh; k += 128) {
    float4 a = *(const float4*)(wh + k); float4 v = *(const float4*)(h0 + k);
    s = fmaf(a.x, v.x, fmaf(a.y, v.y, fmaf(a.z, v.z, fmaf(a.w, v.w, s))));
  }
  s = wredSum(s);
  if (lane == 0) gates[row] = s + bih[row] + bhh[row];
}

__global__ void k_lstm(const float* __restrict__ gates, const float* __restrict__ c0,
                       float* __restrict__ h1, float* __restrict__ c1,
                       float* __restrict__ out_h, float* __restrict__ out_c) {
  int i = blockIdx.x * blockDim.x + threadIdx.x;
  if (i >= Hh) return;
  float ig = 1.f / (1.f + __expf(-gates[i]));
  float fg = 1.f / (1.f + __expf(-gates[Hh + i]));
  float gg = tanhf(gates[2 * Hh + i]);
  float og = 1.f / (1.f + __expf(-gates[3 * Hh + i]));
  float c = fmaf(fg, c0[i], ig * gg);
  float h = og * tanhf(c);
  c1[i] = c; h1[i] = h; out_h[i] = h; out_c[i] = c;
}

// gen_p = sigmoid( W_gen . concat(h1, name_h, type_h, x) + b_gen )
__global__ void k_gen(const float* __restrict__ Wg, const float* __restrict__ bg,
                      const float* __restrict__ h1, const float* __restrict__ nh,
                      const float* __restrict__ th, const float* __restrict__ x,
                      float* __restrict__ gen_p) {
  int tid = threadIdx.x;
  float s = 0.f;
  for (int k = tid; k < COMB; k += blockDim.x) {
    float cv = (k < Hh) ? h1[k]
             : (k < 2 * Hh) ? nh[k - Hh]
             : (k < 3 * Hh) ? th[k - 2 * Hh]
             : x[k - 3 * Hh];
    s = fmaf(Wg[k], cv, s);
  }
  s = blockSum(s);
  if (tid == 0) *gen_p = 1.f / (1.f + __expf(-(s + bg[0])));
}

// per-s row softmax of attn scores, weight by sim[s], scatter-add into p_copy
__global__ void k_attn_scatter(const float* __restrict__ scores, const float* __restrict__ sim,
                               const int* __restrict__ idx, float* __restrict__ pcopy) {
  int s = blockIdx.x;           // 0..7
  int t = threadIdx.x;          // 0..127
  float val = scores[s * TSs + t];
  float m = blockMax(val);
  float e = __expf(val - m);
  float sum = blockSum(e);
  float w = sim[s] * e / sum;
  atomicAdd(&pcopy[idx[s * TSs + t]], w);
}

// ---------------- Big vocab GEMV via fp32 WMMA, double-buffered LDS ----------------
// Block: 256 threads = 8 waves; 16 rows/wave -> 128 rows/block.
// Pipeline per 32-k tile: issue global_load_b128 for tile t+1 into regs,
// run 8x V_WMMA_F32_16X16X4_F32 on tile t from LDS, barrier, commit regs to the
// alternate LDS buffer, barrier.  A[m,k] = h[k0+k] (all M rows identical),
// B[k,n] = W[rowBase+n, k0+k] => every D row = the 16 outputs; lane n<16 reads acc[0].
#define WOUT_KC 32
#define WOUT_PAD (WOUT_KC + 4)   // 36-float row pitch: 16B-aligned float4 stores,
                                 // conflict-free lane-half reads (36*dL != 2 mod 64)
__global__ __launch_bounds__(256) void k_wout(const float* __restrict__ W,
                                              const float* __restrict__ h,
                                              const float* __restrict__ bias,
                                              float* __restrict__ logits,
                                              int nrows, int K) {
  __shared__ float sW[2][128 * WOUT_PAD];
  __shared__ float sHv[Hh];
  const int tid  = threadIdx.x;
  const int wave = tid >> 5;
  const int lane = tid & 31;
  const int rowBase = blockIdx.x * 128;

  // preload entire h vector into LDS once (K <= Hh, multiple of 4*blockDim)
  for (int k = tid * 4; k < K; k += blockDim.x * 4)
    *(float4*)&sHv[k] = *(const float4*)(h + k);

  const int r       = tid >> 1;          // 0..127 : row staged by this thread
  const int cbase   = (tid & 1) * 16;    // 0 or 16
  const int grow_ld = rowBase + r;
  const bool valid  = (grow_ld < nrows);
  const float* srcBase = W + (long long)grow_ld * K + cbase;

  float4 rg[4] = {make_float4(0.f, 0.f, 0.f, 0.f), make_float4(0.f, 0.f, 0.f, 0.f),
                  make_float4(0.f, 0.f, 0.f, 0.f), make_float4(0.f, 0.f, 0.f, 0.f)};
  // prologue: fetch tile 0 and commit to buffer 0
  if (valid) {
#pragma unroll
    for (int c = 0; c < 4; ++c) rg[c] = *(const float4*)(srcBase + c * 4);
  }
#pragma unroll
  for (int c = 0; c < 4; ++c)
    *(float4*)&sW[0][r * WOUT_PAD + cbase + c * 4] = rg[c];
  __syncthreads();

  v8f acc = {};
  const int rloc = wave * 16 + (lane & 15);   // N-column = W row within block
  const int hi   = (lane >> 4) * 2;           // lane-half K offset: 0 or 2
  const int nTiles = K / WOUT_KC;

  for (int t = 0; t < nTiles; ++t) {
    const int k0  = t * WOUT_KC;
    const int cur = t & 1;
    const bool haveNext = (t + 1) < nTiles;

    // issue next tile's global loads before computing (latency overlap)
    if (haveNext && valid) {
#pragma unroll
      for (int c = 0; c < 4; ++c)
        rg[c] = *(const float4*)(srcBase + k0 + WOUT_KC + c * 4);
      if (t + 2 < nTiles)
        __builtin_prefetch(srcBase + k0 + 2 * WOUT_KC, 0, 0);  // global_prefetch_b8
    }

    // compute tile t from LDS
    const float* bufp = &sW[cur][rloc * WOUT_PAD];
    const float* hp   = &sHv[k0];
#pragma unroll
    for (int kk = 0; kk < WOUT_KC; kk += 4) {
      v2f a, b;
      a.x = hp[kk + hi + 0];
      a.y = hp[kk + hi + 1];
      b.x = bufp[kk + hi + 0];
      b.y = bufp[kk + hi + 1];
      acc = __builtin_amdgcn_wmma_f32_16x16x4_f32(
          /*neg_a=*/false, a, /*neg_b=*/false, b,
          /*c_mod=*/(short)0, acc, /*reuse_a=*/false, /*reuse_b=*/false);
    }
    __syncthreads();                       // everyone done reading sW[cur]
    if (haveNext) {
#pragma unroll
      for (int c = 0; c < 4; ++c)
        *(float4*)&sW[1 - cur][r * WOUT_PAD + cbase + c * 4] = rg[c];
      __syncthreads();                     // next tile visible before its compute
    }
  }

  // D[M=0, N=lane] lives in acc[0] for lanes 0..15
  const int n = lane & 15;
  const int grow = rowBase + wave * 16 + n;
  if (lane < 16 && grow < nrows) logits[grow] = acc[0] + bias[grow];
}

// single block: max + inv-sum-exp over logits
__global__ void k_vreduce(const float* __restrict__ logits, int n, float* __restrict__ ms) {
  int tid = threadIdx.x;
  float m = -INFINITY;
  for (int k = tid; k < n; k += blockDim.x) m = fmaxf(m, logits[k]);
  m = blockMax(m);
  float s = 0.f;
  for (int k = tid; k < n; k += blockDim.x) s += __expf(logits[k] - m);
  s = blockSum(s);
  if (tid == 0) { ms[0] = m; ms[1] = 1.f / s; }
}

// out[i] = log(clip(gen_p * p_vocab[i] + (1-gen_p) * p_copy[i], 1e-10))
__global__ void k_final(const float* __restrict__ logits, const float* __restrict__ ms,
                        const float* __restrict__ pcopy, const float* __restrict__ gen_p,
                        float* __restrict__ out, int nvocab, int total) {
  int i = blockIdx.x * blockDim.x + threadIdx.x;
  if (i >= total) return;
  float gp = *gen_p;
  float pv = (i < nvocab) ? __expf(logits[i] - ms[0]) * ms[1] : 0.f;
  float p = fmaf(gp, pv, (1.f - gp) * pcopy[i]);
  out[i] = __logf(fmaxf(p, 1e-10f));
}

// ---------------- Launch ----------------
extern "C" void kernel_launch(void* const* d_in, const int* in_sizes, int n_in,
                              void* d_out, int out_size, void* d_ws, size_t ws_size,
                              hipStream_t stream) {
  (void)in_sizes; (void)n_in; (void)out_size; (void)ws_size;
  const float* input_emb = (const float*)d_in[0];
  const float* h0        = (const float*)d_in[1];
  const float* c0        = (const float*)d_in[2];
  const float* enc       = (const float*)d_in[3];
  const float* name_hid  = (const float*)d_in[4];
  const float* type_hid  = (const float*)d_in[5];
  const float* name_h    = (const float*)d_in[6];
  const float* type_h    = (const float*)d_in[7];
  const float* W_ma_attn = (const float*)d_in[8];
  const float* b_ma_attn = (const float*)d_in[9];
  const float* W_ma_comb = (const float*)d_in[10];
  const float* b_ma_comb = (const float*)d_in[11];
  const float* W_ih      = (const float*)d_in[12];
  const float* W_hh      = (const float*)d_in[13];
  const float* b_ih      = (const float*)d_in[14];
  const float* b_hh      = (const float*)d_in[15];
  const float* W_copy    = (const float*)d_in[16];
  const float* b_copy    = (const float*)d_in[17];
  const float* W_gen     = (const float*)d_in[18];
  const float* b_gen     = (const float*)d_in[19];
  const float* W_out     = (const float*)d_in[20];
  const float* b_out     = (const float*)d_in[21];
  const int*   type_idx  = (const int*)d_in[22];
  float* out = (float*)d_out;
  float* ws  = (float*)d_ws;

  // p_copy must be zero each call (harness poisons ws once, never re-poisons)
  k_zero<<<(VPO + 255) / 256, 256, 0, stream>>>(ws + WS_PCOPY, VPO);

  // q = W_ma_attn @ e + b        (1024 x 512)
  k_gemv<<<128, 256, 0, stream>>>(W_ma_attn, input_emb, b_ma_attn, ws + WS_Q, Hh, Ee, Ee, 0);

  // attention scores
  k_gemv<<<64, 256, 0, stream>>>(enc,      ws + WS_Q, nullptr, ws + WS_SC_ENC,  Tt,      Hh, Hh, 0);
  k_gemv<<<128, 256, 0, stream>>>(name_hid, ws + WS_Q, nullptr, ws + WS_SC_NAME, Ss * TSs, Hh, Hh, 0);
  k_gemv<<<128, 256, 0, stream>>>(type_hid, ws + WS_Q, nullptr, ws + WS_SC_TYPE, Ss * TSs, Hh, Hh, 0);
  k_softmax<<<1, 1024, 0, stream>>>(ws + WS_SC_ENC, Tt);
  k_softmax<<<1, 1024, 0, stream>>>(ws + WS_SC_NAME, Ss * TSs);
  k_softmax<<<1, 1024, 0, stream>>>(ws + WS_SC_TYPE, Ss * TSs);

  // contexts
  k_colsum<<<4, 256, 0, stream>>>(enc,      ws + WS_SC_ENC,  ws + WS_C_ENC,  Tt,      Hh);
  k_colsum<<<4, 256, 0, stream>>>(name_hid, ws + WS_SC_NAME, ws + WS_C_NAME, Ss * TSs, Hh);
  k_colsum<<<4, 256, 0, stream>>>(type_hid, ws + WS_SC_TYPE, ws + WS_C_TYPE, Ss * TSs, Hh);

  // x = tanh(W_ma_comb @ comb + b)
  k_x<<<64, 256, 0, stream>>>(W_ma_comb, b_ma_comb, input_emb,
                              ws + WS_C_ENC, ws + WS_C_NAME, ws + WS_C_TYPE, ws + WS_X);

  // LSTM
  k_gates<<<512, 256, 0, stream>>>(W_ih, W_hh, b_ih, b_hh, ws + WS_X, h0, ws + WS_GATES);
  k_lstm<<<4, 256, 0, stream>>>(ws + WS_GATES, c0, ws + WS_H1, ws + WS_C1,
                                out + VPO, out + VPO + Hh);

  // copy path
  k_gemv<<<128, 256, 0, stream>>>(W_copy, ws + WS_H1, b_copy, ws + WS_QC, Hh, Hh, Hh, 0);
  k_gemv<<<128, 256, 0, stream>>>(type_hid, ws + WS_QC, nullptr, ws + WS_ATTN, Ss * TSs, Hh, Hh, 0);
  // sim scores: rows = name_hiddens[s, TS-1, :], x = encoder_outputs[0, T-1, :]
  k_gemv<<<1, 256, 0, stream>>>(name_hid, enc + (long long)(Tt - 1) * Hh, nullptr, ws + WS_SIM,
                                Ss, Hh, (long long)TSs * Hh, (long long)(TSs - 1) * Hh);
  k_gen<<<1, 256, 0, stream>>>(W_gen, b_gen, ws + WS_H1, name_h, type_h, ws + WS_X, ws + WS_GENP);
  k_softmax<<<1, 1024, 0, stream>>>(ws + WS_SIM, Ss);
  k_attn_scatter<<<Ss, TSs, 0, stream>>>(ws + WS_ATTN, ws + WS_SIM, type_idx, ws + WS_PCOPY);

  // vocab logits via WMMA (dominant: 206 MB stream of W_out)
  k_wout<<<(Vv + 127) / 128, 256, 0, stream>>>(W_out, ws + WS_H1, b_out, ws + WS_LOGITS, Vv, Hh);
  k_vreduce<<<1, 1024, 0, stream>>>(ws + WS_LOGITS, Vv, ws + WS_MS);

  // final combine + log
  k_final<<<(VPO + 255) / 256, 256, 0, stream>>>(ws + WS_LOGITS, ws + WS_MS,
                                                 ws + WS_PCOPY, ws + WS_GENP, out, Vv, VPO);
}